// QuadROI_77515569758786
// MI455X (gfx1250) — compile-verified
//
#include <hip/hip_runtime.h>
#include <hip/hip_bf16.h>

typedef __attribute__((ext_vector_type(2))) float v2f;
typedef __attribute__((ext_vector_type(8))) float v8f;

#define IMG_H 1024
#define IMG_W 1024
#define CH    3
#define ROI_H 32
#define ROI_W 256
#define ROWS_PER_BLK 16   // each block handles one 16-row half of a ROI

// One block = 256 threads (8 waves, wave32) per (box, half).
// Phase 1: WMMA f32 16x16x4 computes the 16x256 (x,y) coordinate grid into LDS.
// Phase 2: bilinear gather (zero padding) + coalesced stores.
__global__ __launch_bounds__(256) void QuadROI_kernel(const float* __restrict__ img,
                                                      const float* __restrict__ boxes,
                                                      float* __restrict__ out) {
    __shared__ float2 xy[ROWS_PER_BLK * ROI_W];   // 32 KB

    const int blk   = blockIdx.x;       // 0..511
    const int boxId = blk >> 1;         // 0..255  (b*64 + n)
    const int th    = blk & 1;          // which 16-row half
    const int tid   = threadIdx.x;      // 0..255
    const int lane  = tid & 31;
    const int wv    = tid >> 5;         // 0..7
    const int b     = boxId >> 6;       // batch index (64 boxes per image)

    // Quad corners (TL,TR,BR,BL), with the grid_sample -0.5 shift folded in:
    // sample coord = grid_pixel_coord - 0.5 for align_corners=False.
    const float* bp = boxes + boxId * 8;
    const float p0x = bp[0] - 0.5f, p0y = bp[1] - 0.5f;
    const float p1x = bp[2] - 0.5f, p1y = bp[3] - 0.5f;
    const float p2x = bp[4] - 0.5f, p2y = bp[5] - 0.5f;
    const float p3x = bp[6] - 0.5f, p3y = bp[7] - 0.5f;

    // ---------------- Phase 1: coordinate grid via V_WMMA_F32_16X16X4_F32 -------------
    // Tile: D(16x16) = A(16x4) x B(4x16), K=2..3 zero-padded.
    //   A row m (output row): [leftX(t), rightX(t), 0, 0],  t = (th*16+m)/31
    //   B col n (output col): [1-s, s, 0, 0]^T,             s = (tw*16+n)/255
    // A layout: lanes 0-15 -> {K0,K1} in vgpr{0,1}; lanes 16-31 -> {K2,K3} (zeros).
    // B layout: vgpr0 = rows K0 (lanes 0-15) / K2 (lanes 16-31); vgpr1 = K1 / K3.
    const int  m  = lane & 15;
    const bool lo = lane < 16;
    const float t     = (float)(th * 16 + m) * (1.0f / 31.0f);
    const float leftX = p0x + t * (p3x - p0x);
    const float rightX= p1x + t * (p2x - p1x);
    const float leftY = p0y + t * (p3y - p0y);
    const float rightY= p1y + t * (p2y - p1y);

    v2f ax, ay;
    ax.x = lo ? leftX  : 0.0f;  ax.y = lo ? rightX : 0.0f;
    ay.x = lo ? leftY  : 0.0f;  ay.y = lo ? rightY : 0.0f;

    #pragma unroll
    for (int i = 0; i < 2; ++i) {
        const int tw   = wv * 2 + i;            // column tile 0..15
        const float s  = (float)(tw * 16 + m) * (1.0f / 255.0f);
        v2f bm;
        bm.x = lo ? (1.0f - s) : 0.0f;          // K=0 row (K=2 zero)
        bm.y = lo ? s          : 0.0f;          // K=1 row (K=3 zero)

        v8f cz = {};
        v8f dx = __builtin_amdgcn_wmma_f32_16x16x4_f32(
                     false, ax, false, bm, (short)0, cz, false, false);
        v8f dy = __builtin_amdgcn_wmma_f32_16x16x4_f32(
                     false, ay, false, bm, (short)0, cz, false, false);

        // C/D layout: vgpr r -> M = r (lanes 0-15) / r+8 (lanes 16-31), N = lane%16
        const int wcol  = tw * 16 + m;
        const int rbase = lo ? 0 : 8;
        #pragma unroll
        for (int r = 0; r < 8; ++r) {
            xy[(rbase + r) * ROI_W + wcol] = make_float2(dx[r], dy[r]);
        }
    }
    __syncthreads();

    // ---------------- Phase 2: bilinear gather (zeros padding) ------------------------
    const float* imgb = img + (size_t)b * (CH * IMG_H * IMG_W);
    float*       outb = out + (size_t)boxId * (CH * ROI_H * ROI_W);

    #pragma unroll 2
    for (int it = 0; it < ROWS_PER_BLK; ++it) {
        const int h = th * 16 + it;             // global ROI row
        const int w = tid;                      // one column per thread
        const float2 g = xy[it * ROI_W + w];

        const float xf = floorf(g.x), yf = floorf(g.y);
        const int x0 = (int)xf, y0 = (int)yf;
        const int x1 = x0 + 1,  y1 = y0 + 1;
        const float wx1 = g.x - xf, wy1 = g.y - yf;
        const float wx0 = 1.0f - wx1, wy0 = 1.0f - wy1;

        const bool vx0 = (x0 >= 0) & (x0 < IMG_W);
        const bool vx1 = (x1 >= 0) & (x1 < IMG_W);
        const bool vy0 = (y0 >= 0) & (y0 < IMG_H);
        const bool vy1 = (y1 >= 0) & (y1 < IMG_H);

        const int xc0 = min(max(x0, 0), IMG_W - 1);
        const int xc1 = min(max(x1, 0), IMG_W - 1);
        const int yc0 = min(max(y0, 0), IMG_H - 1);
        const int yc1 = min(max(y1, 0), IMG_H - 1);

        const float w00 = wy0 * wx0 * ((vy0 & vx0) ? 1.0f : 0.0f);
        const float w01 = wy0 * wx1 * ((vy0 & vx1) ? 1.0f : 0.0f);
        const float w10 = wy1 * wx0 * ((vy1 & vx0) ? 1.0f : 0.0f);
        const float w11 = wy1 * wx1 * ((vy1 & vx1) ? 1.0f : 0.0f);

        const size_t o00 = (size_t)yc0 * IMG_W + xc0;
        const size_t o01 = (size_t)yc0 * IMG_W + xc1;
        const size_t o10 = (size_t)yc1 * IMG_W + xc0;
        const size_t o11 = (size_t)yc1 * IMG_W + xc1;

        #pragma unroll
        for (int c = 0; c < CH; ++c) {
            const float* ic = imgb + (size_t)c * (IMG_H * IMG_W);
            const float acc = ic[o00] * w00 + ic[o01] * w01 +
                              ic[o10] * w10 + ic[o11] * w11;
            outb[c * (ROI_H * ROI_W) + h * ROI_W + w] = acc;
        }
    }
}

extern "C" void kernel_launch(void* const* d_in, const int* in_sizes, int n_in,
                              void* d_out, int out_size, void* d_ws, size_t ws_size,
                              hipStream_t stream) {
    const float* img   = (const float*)d_in[0];   // [4,3,1024,1024] fp32
    const float* boxes = (const float*)d_in[1];   // [4,64,4,2] fp32
    float* out = (float*)d_out;                   // [256,3,32,256] fp32
    (void)in_sizes; (void)n_in; (void)out_size; (void)d_ws; (void)ws_size;

    // 256 boxes x 2 half-grids, 256 threads (8 wave32) each.
    QuadROI_kernel<<<dim3(512), dim3(256), 0, stream>>>(img, boxes, out);
}